// Hough_Transform_6914897347200
// MI455X (gfx1250) — compile-verified
//
#include <hip/hip_runtime.h>
#include <math.h>

#define HH 512
#define WW 512
#define NPIX (HH * WW)

#define T_BINS 180
#define R_BINS 360
#define TPAD 192   // 12 tiles of 16 (bins >= 180 forced to zero weight)
#define RPAD 384   // 3 rho groups x 128 bins (8 tiles); bins >= 360 zero weight
#define BR 128     // rho bins per block
#define NTT 12     // theta tiles per block (all of TPAD)

#define KSPLIT 128
#define PIX_PER_BLOCK (NPIX / KSPLIT)   // 2048
#define KC 16
#define NCHUNK (PIX_PER_BLOCK / KC)     // 128

typedef __attribute__((ext_vector_type(2))) float v2f;
typedef __attribute__((ext_vector_type(8))) float v8f;

constexpr float kThetaMin = -1.5707963267948966f;
constexpr float kTBin     = (float)(3.14159265358979323846 / 179.0);
constexpr float kRhoMin   = -724.0773439350246f;            // -512*sqrt(2)
constexpr float kRBin     = (float)(2.0 * 724.0773439350246 / 359.0);
constexpr float kInvSqrt2 = 0.7071067811865476f;

__global__ __launch_bounds__(256) void hough_zero(float* __restrict__ out, int n) {
  int i = blockIdx.x * 256 + threadIdx.x;
  if (i < n) out[i] = 0.0f;
}

// ---- Phase 1: per-pixel plane fit + polar params (memory-bound stencil) ----
__global__ __launch_bounds__(256) void hough_plane_fit(
    const float* __restrict__ img, const float* __restrict__ mask,
    float* __restrict__ th, float* __restrict__ rh,
    float* __restrict__ it, float* __restrict__ ir,
    float* __restrict__ vf) {
  const int p  = blockIdx.x * 256 + threadIdx.x;
  const int hi = p >> 9;
  const int wi = p & (WW - 1);

  float s[3][3];
#pragma unroll
  for (int dx = -1; dx <= 1; ++dx) {
    const int ri = min(max(hi + dx, 0), HH - 1);
#pragma unroll
    for (int dy = -1; dy <= 1; ++dy) {
      const int ci = min(max(wi + dy, 0), WW - 1);
      s[dx + 1][dy + 1] = img[ri * WW + ci];
    }
  }

  float alpha = 0.f, beta = 0.f, gamma = 0.f;
#pragma unroll
  for (int dx = -1; dx <= 1; ++dx)
#pragma unroll
    for (int dy = -1; dy <= 1; ++dy) {
      const float v = s[dx + 1][dy + 1];
      alpha += (float)dx * v;
      beta  += (float)dy * v;
      gamma += v;
    }
  alpha = alpha / 6.0f + 1e-6f;   // sum_x2 = 6
  beta  = beta  / 6.0f + 1e-6f;   // sum_y2 = 6
  gamma = gamma / 9.0f;

  // faithful to reference: residual uses img[clip(hi+dx), wi] (dy=0 column)
  float eps2 = 0.f;
#pragma unroll
  for (int dx = -1; dx <= 1; ++dx) {
    const float rs = s[dx + 1][1];
#pragma unroll
    for (int dy = -1; dy <= 1; ++dy) {
      const float r = rs - alpha * (float)dy - beta * (float)dx - gamma;
      eps2 += r * r;
    }
  }
  const float noise_var = eps2 / 7.0f;     // n_win - 2
  const float var_ab    = noise_var / 6.0f;

  const float theta = atanf(beta / alpha);
  const float ct = cosf(theta);
  const float st = sinf(theta);
  const float x = (float)hi, y = (float)wi;
  const float rho = x * ct + y * st;
  const float g2  = alpha * alpha + beta * beta;
  const float var_theta = (beta * beta * var_ab + alpha * alpha * var_ab) / (g2 * g2);
  const float drdt = -x * st + y * ct;
  const float var_rho = drdt * drdt * var_theta;

  const float valid = (var_theta <= 100.0f && var_rho <= 100.0f) ? mask[p] : 0.0f;

  th[p] = theta;
  rh[p] = rho;
  it[p] = kInvSqrt2 / sqrtf(var_theta + 1e-12f);  // z = (edge - mu) * it
  ir[p] = kInvSqrt2 / sqrtf(var_rho + 1e-12f);
  vf[p] = valid;
}

// ---- Phase 2: fused Gaussian weights + f32 WMMA outer-product accumulation ----
// grid = (3 rho-groups of 128 bins) x (KSPLIT pixel splits); 256 threads = 8 waves.
// Block covers ALL 192 (padded) theta bins x 128 rho bins (square-ish blocking
// minimizes edge-CDF erf work). Wave w owns rho tile w x all 12 theta tiles.
//
// K-pair interleaving: WMMA A fragments need rows (kb, kb+1) with kb always
// even, so weights/CDFs are stored as float2 {row kb, row kb+1} -> each A
// fragment is a single aligned ds_load_b64, no repack movs.
__global__ __launch_bounds__(256) void hough_vote_wmma(
    const float* __restrict__ th, const float* __restrict__ rh,
    const float* __restrict__ it, const float* __restrict__ ir,
    const float* __restrict__ vf, float* __restrict__ out) {
  __shared__ float cdfT[KC][194];    // 193 theta edges per pixel (stride 194)
  __shared__ v2f   cdfRP[KC / 2][130]; // rho edge CDFs, k-pair interleaved
  __shared__ v2f   wtP[KC / 2][TPAD];  // theta weights, k-pair interleaved
  __shared__ float thS[KC], rhS[KC], itS[KC], irS[KC], vfS[KC];

  const int tid  = threadIdx.x;
  const int lane = tid & 31;
  const int wave = tid >> 5;
  const int rg   = blockIdx.x;                 // rho group: bins [rg*128, rg*128+128)
  const int p0   = blockIdx.y * PIX_PER_BLOCK;
  const int m15  = lane & 15;
  const int k2of = lane >> 4;                  // k-pair offset within k-step
  const int bcol = wave * 16 + m15;            // this wave's rho column (local)

  v8f acc[NTT];
#pragma unroll
  for (int t = 0; t < NTT; ++t) acc[t] = {};

  for (int chunk = 0; chunk < NCHUNK; ++chunk) {
    const int pb = p0 + chunk * KC;
    if (tid < KC) {
      const int p = pb + tid;
      thS[tid] = th[p];
      rhS[tid] = rh[p];
      itS[tid] = it[p];
      irS[tid] = ir[p];
      vfS[tid] = vf[p];
    }
    __syncthreads();

    // ---- theta edge CDFs: KC x 193 = 3088 entries, one erf each ----
#pragma unroll
    for (int i = 0; i < 13; ++i) {
      const int idx = i * 256 + tid;
      if (idx < KC * 193) {
        const int k = idx / 193;
        const int j = idx - k * 193;
        const float e = kThetaMin + ((float)j - 0.5f) * kTBin;
        cdfT[k][j] = erff((e - thS[k]) * itS[k]);
      }
    }
    // ---- rho edge CDFs (k-pair interleaved): 8 x 129 = 1032 float2 entries.
    // Edge index clamps at 360 so padded bins (>= 360) difference to zero.
#pragma unroll
    for (int i = 0; i < 5; ++i) {
      const int idx = i * 256 + tid;
      if (idx < (KC / 2) * 129) {
        const int k2 = idx / 129;
        const int r  = idx - k2 * 129;
        const int ge = min(rg * BR + r, R_BINS);
        const float e = kRhoMin + ((float)ge - 0.5f) * kRBin;
        v2f c;
        c.x = erff((e - rhS[2 * k2]) * irS[2 * k2]);
        c.y = erff((e - rhS[2 * k2 + 1]) * irS[2 * k2 + 1]);
        cdfRP[k2][r] = c;
      }
    }
    __syncthreads();

    // ---- theta weights (k-pair interleaved): 8 x 192 = 1536 float2 entries ----
#pragma unroll
    for (int i = 0; i < ((KC / 2) * TPAD) / 256; ++i) {
      const int idx = i * 256 + tid;
      const int k2 = idx / TPAD;
      const int m  = idx - k2 * TPAD;
      v2f w = {};
      if (m < T_BINS) {
        w.x = 0.5f * (cdfT[2 * k2][m + 1] - cdfT[2 * k2][m]) * vfS[2 * k2];
        w.y = 0.5f * (cdfT[2 * k2 + 1][m + 1] - cdfT[2 * k2 + 1][m]) * vfS[2 * k2 + 1];
      }
      wtP[k2][m] = w;
    }
    __syncthreads();

    // ---- 16 pixels = 4 k-steps of V_WMMA_F32_16X16X4_F32 ----
    // One B fragment per k-step (packed diff), reused by 12 WMMAs;
    // each A fragment is a single b64 LDS load.
#pragma unroll
    for (int ks = 0; ks < 4; ++ks) {
      const int k2 = ks * 2 + k2of;
      const v2f b = cdfRP[k2][bcol + 1] - cdfRP[k2][bcol];
#pragma unroll
      for (int t = 0; t < NTT; ++t) {
        const v2f a = wtP[k2][t * 16 + m15];
        acc[t] = __builtin_amdgcn_wmma_f32_16x16x4_f32(
            false, a, false, b, (short)0, acc[t], false, false);
      }
    }
    __syncthreads();
  }

  // epilogue: C/D layout -> element (m,n) in vgpr m%8, lane (m/8)*16+n
  const int mhi = (lane >> 4) * 8;
#pragma unroll
  for (int t = 0; t < NTT; ++t) {
#pragma unroll
    for (int v = 0; v < 8; ++v) {
      const int gt = t * 16 + mhi + v;
      const int gr = rg * BR + wave * 16 + m15;
      if (gt < T_BINS && gr < R_BINS)
        atomicAdd(&out[gt * R_BINS + gr], acc[t][v]);
    }
  }
}

extern "C" void kernel_launch(void* const* d_in, const int* in_sizes, int n_in,
                              void* d_out, int out_size, void* d_ws, size_t ws_size,
                              hipStream_t stream) {
  (void)in_sizes; (void)n_in; (void)out_size; (void)ws_size;
  const float* img  = (const float*)d_in[0];
  const float* mask = (const float*)d_in[1];
  float* out = (float*)d_out;
  float* ws  = (float*)d_ws;
  float* th = ws;
  float* rh = ws + (size_t)NPIX;
  float* it = ws + 2 * (size_t)NPIX;
  float* ir = ws + 3 * (size_t)NPIX;
  float* vf = ws + 4 * (size_t)NPIX;

  hough_zero<<<(T_BINS * R_BINS + 255) / 256, 256, 0, stream>>>(out, T_BINS * R_BINS);
  hough_plane_fit<<<NPIX / 256, 256, 0, stream>>>(img, mask, th, rh, it, ir, vf);
  dim3 grid(RPAD / BR, KSPLIT);
  hough_vote_wmma<<<grid, 256, 0, stream>>>(th, rh, it, ir, vf, out);
}